// GCN_35897336660004
// MI455X (gfx1250) — compile-verified
//
#include <hip/hip_runtime.h>
#include <math.h>

typedef __attribute__((ext_vector_type(2))) float v2f;
typedef __attribute__((ext_vector_type(8))) float v8f;

#define N_NODES 50000
#define N_EDGES 800000
#define NFEAT 256
#define NHID 64
#define NCLASS 40

#define NT1 4                    // 64/16 column tiles
#define TILES1 ((N_NODES/16)*NT1)   // 3125*4 = 12500
#define NT2 3                    // ceil(40/16) column tiles
#define TILES2 ((N_NODES/16)*NT2)   // 3125*3 = 9375

// ---------------------------------------------------------------------------
// Zero the SpMM accumulators (acc1 in ws, logits region of d_out).
__global__ __launch_bounds__(256) void zero2_kernel(float* __restrict__ a, int na,
                                                    float* __restrict__ b, int nb) {
    int i = blockIdx.x * blockDim.x + threadIdx.x;
    if (i < na) a[i] = 0.0f;
    if (i < nb) b[i] = 0.0f;
}

// ---------------------------------------------------------------------------
// GEMM1: t1[50000,64] = x[50000,256] @ W1[256,64], f32 WMMA 16x16x4.
// One wave per 16x16 output tile.
__global__ __launch_bounds__(256) void gemm1_kernel(const float* __restrict__ x,
                                                    const float* __restrict__ W1,
                                                    float* __restrict__ t1) {
    int wid  = (blockIdx.x * blockDim.x + threadIdx.x) >> 5;
    if (wid >= TILES1) return;              // uniform per wave -> EXEC all-1s inside
    int lane = threadIdx.x & 31;
    int half = lane >> 4;                   // 0: K=k,k+1   1: K=k+2,k+3
    int lid  = lane & 15;
    int mt = wid / NT1, nt = wid % NT1;
    int m0 = mt * 16, n0 = nt * 16;

    const float* arow = x  + (size_t)(m0 + lid) * NFEAT;
    const float* bcol = W1 + n0 + lid;

    v8f c = {};
    for (int k = 0; k < NFEAT; k += 4) {
        float2 a2 = *(const float2*)(arow + k + 2 * half);   // contiguous, 8B aligned
        v2f a; a.x = a2.x; a.y = a2.y;
        v2f b;
        b.x = bcol[(size_t)(k + 2 * half)     * NHID];
        b.y = bcol[(size_t)(k + 2 * half + 1) * NHID];
        c = __builtin_amdgcn_wmma_f32_16x16x4_f32(false, a, false, b,
                                                  (short)0, c, false, false);
    }
    // D layout: c[r] is (row m0 + r + 8*half, col n0 + lid)
    float* drow = t1 + (size_t)(m0 + 8 * half) * NHID + n0 + lid;
#pragma unroll
    for (int r = 0; r < 8; ++r)
        drow[(size_t)r * NHID] = c[r];
}

// ---------------------------------------------------------------------------
// SpMM layer 1: acc[dst] += t1[src] * w  (64 feats = 32 lanes x float2)
__global__ __launch_bounds__(256) void spmm1_kernel(const float* __restrict__ t1,
                                                    const int* __restrict__ src,
                                                    const int* __restrict__ dst,
                                                    const float* __restrict__ ew,
                                                    float* __restrict__ acc) {
    int wid  = (blockIdx.x * blockDim.x + threadIdx.x) >> 5;
    if (wid >= N_EDGES) return;
    int lane = threadIdx.x & 31;
    int s = src[wid], d = dst[wid];
    float w = ew[wid];
    float2 v = *(const float2*)(t1 + (size_t)s * NHID + lane * 2);
    atomicAdd(acc + (size_t)d * NHID + lane * 2,     v.x * w);
    atomicAdd(acc + (size_t)d * NHID + lane * 2 + 1, v.y * w);
}

// ---------------------------------------------------------------------------
// h = relu(acc + b1), in place.
__global__ __launch_bounds__(256) void bias_relu_kernel(float* __restrict__ acc,
                                                        const float* __restrict__ b1) {
    int i = blockIdx.x * blockDim.x + threadIdx.x;
    if (i < N_NODES * NHID)
        acc[i] = fmaxf(acc[i] + b1[i & (NHID - 1)], 0.0f);
}

// ---------------------------------------------------------------------------
// GEMM2: t2[50000,40] = h[50000,64] @ W2[64,40], f32 WMMA, padded N tiles.
__global__ __launch_bounds__(256) void gemm2_kernel(const float* __restrict__ h,
                                                    const float* __restrict__ W2,
                                                    float* __restrict__ t2) {
    int wid  = (blockIdx.x * blockDim.x + threadIdx.x) >> 5;
    if (wid >= TILES2) return;
    int lane = threadIdx.x & 31;
    int half = lane >> 4;
    int lid  = lane & 15;
    int mt = wid / NT2, nt = wid % NT2;
    int m0 = mt * 16, n0 = nt * 16;
    int col = n0 + lid;
    bool valid = col < NCLASS;

    const float* arow = h + (size_t)(m0 + lid) * NHID;

    v8f c = {};
    for (int k = 0; k < NHID; k += 4) {
        float2 a2 = *(const float2*)(arow + k + 2 * half);
        v2f a; a.x = a2.x; a.y = a2.y;
        v2f b;
        b.x = valid ? W2[(size_t)(k + 2 * half)     * NCLASS + col] : 0.0f;
        b.y = valid ? W2[(size_t)(k + 2 * half + 1) * NCLASS + col] : 0.0f;
        c = __builtin_amdgcn_wmma_f32_16x16x4_f32(false, a, false, b,
                                                  (short)0, c, false, false);
    }
    if (valid) {
        float* drow = t2 + (size_t)(m0 + 8 * half) * NCLASS + col;
#pragma unroll
        for (int r = 0; r < 8; ++r)
            drow[(size_t)r * NCLASS] = c[r];
    }
}

// ---------------------------------------------------------------------------
// SpMM layer 2: logits[dst] += t2[src] * w  (40 feats = 20 lanes x float2)
__global__ __launch_bounds__(256) void spmm2_kernel(const float* __restrict__ t2,
                                                    const int* __restrict__ src,
                                                    const int* __restrict__ dst,
                                                    const float* __restrict__ ew,
                                                    float* __restrict__ logits) {
    int wid  = (blockIdx.x * blockDim.x + threadIdx.x) >> 5;
    if (wid >= N_EDGES) return;
    int lane = threadIdx.x & 31;
    if (lane >= NCLASS / 2) return;
    int s = src[wid], d = dst[wid];
    float w = ew[wid];
    float2 v = *(const float2*)(t2 + (size_t)s * NCLASS + lane * 2);
    atomicAdd(logits + (size_t)d * NCLASS + lane * 2,     v.x * w);
    atomicAdd(logits + (size_t)d * NCLASS + lane * 2 + 1, v.y * w);
}

// ---------------------------------------------------------------------------
// logits += b2; logp = log_softmax(logits).  One wave per node (wave32).
__global__ __launch_bounds__(256) void bias2_lsm_kernel(float* __restrict__ logits,
                                                        const float* __restrict__ b2,
                                                        float* __restrict__ logp) {
    int wid  = (blockIdx.x * blockDim.x + threadIdx.x) >> 5;
    if (wid >= N_NODES) return;
    int lane = threadIdx.x & 31;
    bool active = lane < NCLASS / 2;   // 20 lanes x 2 values = 40

    float x0 = -INFINITY, x1 = -INFINITY;
    if (active) {
        float2 v = *(const float2*)(logits + (size_t)wid * NCLASS + lane * 2);
        x0 = v.x + b2[lane * 2];
        x1 = v.y + b2[lane * 2 + 1];
        float2 o; o.x = x0; o.y = x1;
        *(float2*)(logits + (size_t)wid * NCLASS + lane * 2) = o;
    }
    float m = fmaxf(x0, x1);
#pragma unroll
    for (int off = 16; off > 0; off >>= 1)
        m = fmaxf(m, __shfl_xor(m, off, 32));
    float s = active ? (expf(x0 - m) + expf(x1 - m)) : 0.0f;
#pragma unroll
    for (int off = 16; off > 0; off >>= 1)
        s += __shfl_xor(s, off, 32);
    float ls = m + logf(s);
    if (active) {
        float2 o; o.x = x0 - ls; o.y = x1 - ls;
        *(float2*)(logp + (size_t)wid * NCLASS + lane * 2) = o;
    }
}

// ---------------------------------------------------------------------------
extern "C" void kernel_launch(void* const* d_in, const int* in_sizes, int n_in,
                              void* d_out, int out_size, void* d_ws, size_t ws_size,
                              hipStream_t stream) {
    const float* x    = (const float*)d_in[0];
    const float* W1   = (const float*)d_in[1];
    const float* b1   = (const float*)d_in[2];
    const float* W2   = (const float*)d_in[3];
    const float* b2   = (const float*)d_in[4];
    const float* ew   = (const float*)d_in[5];
    const int*   esrc = (const int*)d_in[6];
    const int*   edst = (const int*)d_in[7];

    float* logits = (float*)d_out;                       // [50000,40]
    float* logp   = logits + (size_t)N_NODES * NCLASS;   // [50000,40]

    float* t1   = (float*)d_ws;                          // [50000,64]
    float* acc1 = t1   + (size_t)N_NODES * NHID;         // [50000,64]
    float* t2   = acc1 + (size_t)N_NODES * NHID;         // [50000,40]

    const int nH = N_NODES * NHID;    // 3,200,000
    const int nC = N_NODES * NCLASS;  // 2,000,000

    zero2_kernel<<<(nH + 255) / 256, 256, 0, stream>>>(acc1, nH, logits, nC);

    gemm1_kernel<<<(TILES1 * 32 + 255) / 256, 256, 0, stream>>>(x, W1, t1);

    spmm1_kernel<<<(N_EDGES * 32 + 255) / 256, 256, 0, stream>>>(t1, esrc, edst, ew, acc1);

    bias_relu_kernel<<<(nH + 255) / 256, 256, 0, stream>>>(acc1, b1);

    gemm2_kernel<<<(TILES2 * 32 + 255) / 256, 256, 0, stream>>>(acc1, W2, t2);

    spmm2_kernel<<<(N_EDGES * 32 + 255) / 256, 256, 0, stream>>>(t2, esrc, edst, ew, logits);

    bias2_lsm_kernel<<<(N_NODES * 32 + 255) / 256, 256, 0, stream>>>(logits, b2, logp);
}